// Attention_F_12214886990460
// MI455X (gfx1250) — compile-verified
//
#include <hip/hip_runtime.h>
#include <math.h>

// ---------------------------------------------------------------------------
// MI455X (gfx1250) frequency-domain channel attention.
// b=4, c=256, h=w=128, heads=8, c'=32, n=h*w=16384.
// Bandwidth-bound (~1.5GB traffic / 23.3TB/s ≈ 65us floor; only ~30 GFLOP),
// so matrix ops stay in f32 via V_WMMA_F32_16X16X4_F32, and operand staging
// uses the Tensor Data Mover (tensor_load_to_lds + s_wait_tensorcnt) with
// TDM LDS padding to avoid bank conflicts.
// ---------------------------------------------------------------------------

typedef float v2f __attribute__((ext_vector_type(2)));
typedef float v8f __attribute__((ext_vector_type(8)));
typedef unsigned u32x4 __attribute__((ext_vector_type(4)));
typedef int i32x4 __attribute__((ext_vector_type(4)));
typedef int i32x8 __attribute__((ext_vector_type(8)));

#define TWO_PI 6.28318530717958647692f

static constexpr int B   = 4;
static constexpr int C   = 256;
static constexpr int H   = 128;
static constexpr int W   = 128;
static constexpr int N   = H * W;          // 16384
static constexpr int HD  = 8;
static constexpr int CPH = 32;             // channels per head
static constexpr int CR  = 16;             // reduced channels in gating branch
static constexpr size_t PLANE = (size_t)B * C * N;  // 16,777,216 floats = 64MB

__device__ __forceinline__ v8f wmma_f32(v2f a, v2f b, v8f c) {
  return __builtin_amdgcn_wmma_f32_16x16x4_f32(false, a, false, b, (short)0, c,
                                               false, false);
}

// On gfx1250 the LDS flat aperture maps addr[31:0] directly to the LDS
// offset, so the low 32 bits of a generic pointer to __shared__ data are the
// byte address TDM descriptors want.
__device__ __forceinline__ unsigned lds_off(const void* p) {
  return (unsigned)(uintptr_t)p;
}

// ---------------------------------------------------------------------------
// Tensor Data Mover: load a 2D f32 tile (tile_h rows x tile_w elems, global
// row stride row_stride elems) into LDS.  Optional LDS padding of
// (pad_amount+1 DWORDs) after every 2^(pad_interval+1) DWORDs lets us write a
// bank-conflict-free row stride directly from the DMA engine.
// D# field packing per CDNA5 ISA 8.3/8.4.
// ---------------------------------------------------------------------------
__device__ __forceinline__ void tdm_load_tile_f32(
    unsigned lds_byte_off, const float* tile_src, unsigned tensor_w,
    unsigned tensor_h, unsigned row_stride, unsigned tile_w, unsigned tile_h,
    unsigned pad_enable, unsigned pad_interval, unsigned pad_amount) {
  unsigned long long ga = (unsigned long long)(uintptr_t)tile_src;
  u32x4 g0;
  g0.x = 1u;                                   // count=1, user-mode descriptor
  g0.y = lds_byte_off;                         // lds_addr
  g0.z = (unsigned)ga;                         // global_addr[31:0]
  g0.w = (unsigned)((ga >> 32) & 0x01FFFFFFull) | (2u << 30);  // addr hi|type=2
  i32x8 g1;
  g1[0] = (int)((2u << 16) |                   // data_size = 4B
                (pad_enable << 20) | (pad_interval << 22) |
                (pad_amount << 25));           // no multicast / atomics
  g1[1] = (int)((tensor_w & 0xFFFFu) << 16);   // tensor_dim0 lo
  g1[2] = (int)((tensor_w >> 16) | ((tensor_h & 0xFFFFu) << 16));
  g1[3] = (int)((tensor_h >> 16) | ((tile_w & 0xFFFFu) << 16));
  g1[4] = (int)(tile_h & 0xFFFFu);             // tile_dim1 (tile_dim2 = 0)
  g1[5] = (int)row_stride;                     // tensor_dim0_stride lo32
  g1[6] = 0;
  g1[7] = 0;
  i32x4 zz = {0, 0, 0, 0};
#if __clang_major__ >= 23
  i32x8 z8 = {0, 0, 0, 0, 0, 0, 0, 0};
  __builtin_amdgcn_tensor_load_to_lds(g0, g1, zz, zz, z8, 0);
#else
  __builtin_amdgcn_tensor_load_to_lds(g0, g1, zz, zz, 0);
#endif
}

// ---------------------------------------------------------------------------
// 128-point radix-2 DIT FFT in LDS.  64 threads per block, one line per block.
// Rows: lines_per_img=128, inner_mul=128, estride=1.
// Cols: lines_per_img=128, inner_mul=1,   estride=128.
// sign = -1 forward, +1 inverse.  absOut != nullptr -> write |z|*scale there.
// ---------------------------------------------------------------------------
__global__ void __launch_bounds__(64)
fft128_c2c(float* re, float* im, int lines_per_img, int inner_mul, int estride,
           float sign, float scale, float* absOut) {
  __shared__ float sr[128], si[128];
  int line = blockIdx.x;
  int img  = line / lines_per_img;
  int li   = line % lines_per_img;
  size_t base = (size_t)img * N + (size_t)li * inner_mul;
  int t = threadIdx.x;  // 0..63

  for (int e = t; e < 128; e += 64) {
    int r = __brev((unsigned)e) >> 25;  // 7-bit reversal
    sr[r] = re[base + (size_t)e * estride];
    si[r] = im[base + (size_t)e * estride];
  }
  __syncthreads();

  for (int s = 1; s <= 7; ++s) {
    int mh  = 1 << (s - 1);
    int grp = t >> (s - 1);
    int j   = t & (mh - 1);
    int p   = (grp << s) + j;
    float ang = sign * TWO_PI * (float)j / (float)(1 << s);
    float cw, sw;
    __sincosf(ang, &sw, &cw);
    float vr = sr[p + mh], vi = si[p + mh];
    float tr = vr * cw - vi * sw;
    float ti = vr * sw + vi * cw;
    float ur = sr[p], ui = si[p];
    sr[p]      = ur + tr;  si[p]      = ui + ti;
    sr[p + mh] = ur - tr;  si[p + mh] = ui - ti;
    __syncthreads();
  }

  if (absOut) {
    for (int e = t; e < 128; e += 64) {
      float a = sr[e] * scale, bb = si[e] * scale;
      absOut[base + (size_t)e * estride] = sqrtf(a * a + bb * bb);
    }
  } else {
    for (int e = t; e < 128; e += 64) {
      re[base + (size_t)e * estride] = sr[e] * scale;
      im[base + (size_t)e * estride] = si[e] * scale;
    }
  }
}

// Real-to-complex forward 128-pt FFT along rows (w).
__global__ void __launch_bounds__(64)
fft128_rows_r2c(const float* __restrict__ x, float* re, float* im) {
  __shared__ float sr[128], si[128];
  size_t base = (size_t)blockIdx.x * 128;
  int t = threadIdx.x;
  for (int e = t; e < 128; e += 64) {
    int r = __brev((unsigned)e) >> 25;
    sr[r] = x[base + e];
    si[r] = 0.0f;
  }
  __syncthreads();
  for (int s = 1; s <= 7; ++s) {
    int mh  = 1 << (s - 1);
    int grp = t >> (s - 1);
    int j   = t & (mh - 1);
    int p   = (grp << s) + j;
    float ang = -TWO_PI * (float)j / (float)(1 << s);
    float cw, sw;
    __sincosf(ang, &sw, &cw);
    float vr = sr[p + mh], vi = si[p + mh];
    float tr = vr * cw - vi * sw;
    float ti = vr * sw + vi * cw;
    float ur = sr[p], ui = si[p];
    sr[p]      = ur + tr;  si[p]      = ui + ti;
    sr[p + mh] = ur - tr;  si[p + mh] = ui - ti;
    __syncthreads();
  }
  for (int e = t; e < 128; e += 64) {
    re[base + e] = sr[e];
    im[base + e] = si[e];
  }
}

// ---------------------------------------------------------------------------
// Per-channel inverse L2 norm over n: invn[ch] = 1/max(||q_ch||, 1e-12)
// ---------------------------------------------------------------------------
__global__ void __launch_bounds__(256)
norm_kernel(const float* re, const float* im, float* invn) {
  int ch = blockIdx.x;  // 0..B*C-1
  size_t base = (size_t)ch * N;
  float s = 0.0f;
  for (int i = threadIdx.x; i < N; i += 256) {
    float a = re[base + i], b = im[base + i];
    s += a * a + b * b;
  }
  __shared__ float red[256];
  red[threadIdx.x] = s;
  __syncthreads();
  for (int off = 128; off > 0; off >>= 1) {
    if (threadIdx.x < off) red[threadIdx.x] += red[threadIdx.x + off];
    __syncthreads();
  }
  if (threadIdx.x == 0) invn[ch] = 1.0f / fmaxf(sqrtf(red[0]), 1e-12f);
}

// ---------------------------------------------------------------------------
// Gram + complex softmax.  One block per (b,head); 4 waves, each owns one
// 16x16 tile of the 32x32 output.  G = Q Q^T (complex, no conjugate):
//   re(G) = Qr Qr^T - Qi Qi^T,  im(G) = Qr Qi^T + Qi Qr^T
// TDM double-buffers 32x64 f32 tiles of each plane into LDS (row stride 66
// via TDM padding: pad_interval=5 -> every 64 DWORDs, pad_amount=1 -> +2
// DWORDs, killing ds_load bank conflicts); wave 0 drives the DMA and
// overlaps chunk c+1's transfer with chunk c's WMMAs via s_wait_tensorcnt(2).
// ---------------------------------------------------------------------------
static constexpr int KC  = 64;   // K-chunk staged per TDM transfer
static constexpr int QLD = 66;   // padded LDS row stride (floats)

__global__ void __launch_bounds__(128)
gram_attn_kernel(const float* re, const float* im, const float* invn,
                 const float* temperature, float* attn_re, float* attn_im) {
  __shared__ float qtile[2][2][CPH][QLD];  // [buf][re/im][row][col] 33.8KB
  __shared__ float sre[32][32], sim[32][32];

  int bh = blockIdx.x;           // 0..31 (= b*8 + head)
  int head = bh & (HD - 1);
  int c0 = bh * CPH;             // global channel base (b*256 + head*32)
  int t = threadIdx.x;
  int wid = t >> 5, lane = t & 31;
  int mi = wid >> 1, ni = wid & 1;
  int r = lane & 15, half = lane >> 4;
  int koff = 2 * half;

  const float* rebase = re + (size_t)c0 * N;
  const float* imbase = im + (size_t)c0 * N;

  if (wid == 0) {  // prologue: chunk 0 -> buf 0
    tdm_load_tile_f32(lds_off(&qtile[0][0][0][0]), rebase, N, CPH, N, KC, CPH,
                      1, 5, 1);
    tdm_load_tile_f32(lds_off(&qtile[0][1][0][0]), imbase, N, CPH, N, KC, CPH,
                      1, 5, 1);
  }

  v8f crr = {}, cii = {}, cri = {}, cir = {};
  for (int chunk = 0; chunk < N / KC; ++chunk) {
    int buf = chunk & 1;
    if (wid == 0) {
      if (chunk + 1 < N / KC) {
        int nb = buf ^ 1;
        const float* src = rebase + (size_t)(chunk + 1) * KC;
        tdm_load_tile_f32(lds_off(&qtile[nb][0][0][0]), src, N, CPH, N, KC,
                          CPH, 1, 5, 1);
        tdm_load_tile_f32(lds_off(&qtile[nb][1][0][0]),
                          imbase + (size_t)(chunk + 1) * KC, N, CPH, N, KC,
                          CPH, 1, 5, 1);
        __builtin_amdgcn_s_wait_tensorcnt(2);  // current chunk's pair done
      } else {
        __builtin_amdgcn_s_wait_tensorcnt(0);
      }
    }
    __syncthreads();

    const float* tr = &qtile[buf][0][0][0];
    const float* ti = &qtile[buf][1][0][0];
    int arow = (mi * 16 + r) * QLD + koff;
    int brow = (ni * 16 + r) * QLD + koff;
#pragma unroll 4
    for (int kk = 0; kk < KC; kk += 4) {
      v2f ar = *(const v2f*)(tr + arow + kk);
      v2f ai = *(const v2f*)(ti + arow + kk);
      v2f br = *(const v2f*)(tr + brow + kk);
      v2f bi = *(const v2f*)(ti + brow + kk);
      crr = wmma_f32(ar, br, crr);
      cii = wmma_f32(ai, bi, cii);
      cri = wmma_f32(ar, bi, cri);
      cir = wmma_f32(ai, br, cir);
    }
    __syncthreads();  // protect buf before TDM overwrites it in 2 chunks
  }

  float temp = temperature[head];
#pragma unroll
  for (int rr = 0; rr < 8; ++rr) {
    int row = mi * 16 + (half ? rr + 8 : rr);
    int col = ni * 16 + r;
    float scl = invn[c0 + row] * invn[c0 + col] * temp;
    sre[row][col] = (crr[rr] - cii[rr]) * scl;
    sim[row][col] = (cri[rr] + cir[rr]) * scl;
  }
  __syncthreads();

  // complex softmax: softmax(real) rows (threads 0..31), softmax(imag) rows
  // (threads 32..63), each over the 32 columns.
  if (t < 64) {
    int row = t & 31;
    float* rp = (t < 32) ? sre[row] : sim[row];
    float mx = -INFINITY;
    for (int j = 0; j < 32; ++j) mx = fmaxf(mx, rp[j]);
    float sm = 0.0f;
    float e[32];
#pragma unroll
    for (int j = 0; j < 32; ++j) {
      e[j] = __expf(rp[j] - mx);
      sm += e[j];
    }
    float inv = 1.0f / sm;
    float* dst = ((t < 32) ? attn_re : attn_im) + (size_t)bh * 1024 + row * 32;
#pragma unroll
    for (int j = 0; j < 32; ++j) dst[j] = e[j] * inv;
  }
}

// ---------------------------------------------------------------------------
// out = attn @ qkv (complex 32x32 times complex 32x16384 per (b,head)).
// 4 waves per block, each one 16x16 output tile; A (attn) from LDS.
// B is a 32x32 tile per block, staged by TDM (padded row stride 34) instead
// of per-lane stride-64KB dword gathers.  f32 WMMA has no A/B negation, so
// pre-negate Ai for the real part:
//   out_re = Ar*Br + (-Ai)*Bi,  out_im = Ar*Bi + Ai*Br.
// ---------------------------------------------------------------------------
static constexpr int BLD = 34;  // padded LDS row stride for the B tile

__global__ void __launch_bounds__(128)
apply_attn_kernel(const float* re, const float* im, const float* attn_re,
                  const float* attn_im, float* o_re, float* o_im) {
  __shared__ float btile[2][CPH][BLD];  // [re/im][chan][col]  8.7KB
  __shared__ float sare[32][32], saim[32][32];

  int bh = blockIdx.x;           // 0..31
  int t = threadIdx.x;
  int wid = t >> 5, lane = t & 31;
  int mi = wid >> 1, nj = wid & 1;
  int nblk = blockIdx.y * 32;
  int n0 = nblk + nj * 16;
  int r = lane & 15, half = lane >> 4;
  int c0 = bh * CPH;

  if (wid == 0) {  // stage the 32x32 B tile for both planes
    tdm_load_tile_f32(lds_off(&btile[0][0][0]), re + (size_t)c0 * N + nblk, N,
                      CPH, N, 32, CPH, 1, 4, 1);
    tdm_load_tile_f32(lds_off(&btile[1][0][0]), im + (size_t)c0 * N + nblk, N,
                      CPH, N, 32, CPH, 1, 4, 1);
  }
  for (int i = t; i < 1024; i += 128) {
    sare[i >> 5][i & 31] = attn_re[(size_t)bh * 1024 + i];
    saim[i >> 5][i & 31] = attn_im[(size_t)bh * 1024 + i];
  }
  if (wid == 0) __builtin_amdgcn_s_wait_tensorcnt(0);
  __syncthreads();

  int colL = nj * 16 + r;  // column within the staged tile
  v8f cre = {}, cim = {};
#pragma unroll
  for (int kk = 0; kk < 32; kk += 4) {
    int ka = kk + 2 * half;
    v2f ar = {sare[mi * 16 + r][ka], sare[mi * 16 + r][ka + 1]};
    v2f ai = {saim[mi * 16 + r][ka], saim[mi * 16 + r][ka + 1]};
    v2f an = {-ai.x, -ai.y};
    v2f br = {btile[0][ka][colL], btile[0][ka + 1][colL]};
    v2f bi = {btile[1][ka][colL], btile[1][ka + 1][colL]};
    cre = wmma_f32(ar, br, cre);
    cre = wmma_f32(an, bi, cre);
    cim = wmma_f32(ar, bi, cim);
    cim = wmma_f32(ai, br, cim);
  }

#pragma unroll
  for (int rr = 0; rr < 8; ++rr) {
    int row = mi * 16 + (half ? rr + 8 : rr);
    size_t idx = (size_t)(c0 + row) * N + n0 + (lane & 15);
    o_re[idx] = cre[rr];
    o_im[idx] = cim[rr];
  }
}

// ---------------------------------------------------------------------------
// 16384-point inverse FFT along n, fully in LDS (128KB dynamic shared — only
// comfortable with CDNA5's 320KB/WGP).  One block per row; 512 threads,
// 14 radix-2 stages.  Applies the 1/16384 normalization.
// ---------------------------------------------------------------------------
__global__ void __launch_bounds__(512)
ifft16384_rows(float* re, float* im) {
  extern __shared__ float sm[];
  float* sr = sm;
  float* si = sm + N;
  size_t base = (size_t)blockIdx.x * N;
  int t = threadIdx.x;

  for (int e = t; e < N; e += 512) {
    int r = __brev((unsigned)e) >> 18;  // 14-bit reversal
    sr[r] = re[base + e];
    si[r] = im[base + e];
  }
  __syncthreads();

  for (int s = 1; s <= 14; ++s) {
    int mh = 1 << (s - 1);
    for (int bfy = t; bfy < (N >> 1); bfy += 512) {
      int grp = bfy >> (s - 1);
      int j   = bfy & (mh - 1);
      int p   = (grp << s) + j;
      float ang = TWO_PI * (float)j / (float)(1 << s);  // inverse: +
      float cw, sw;
      __sincosf(ang, &sw, &cw);
      float vr = sr[p + mh], vi = si[p + mh];
      float tr = vr * cw - vi * sw;
      float ti = vr * sw + vi * cw;
      float ur = sr[p], ui = si[p];
      sr[p]      = ur + tr;  si[p]      = ui + ti;
      sr[p + mh] = ur - tr;  si[p + mh] = ui - ti;
    }
    __syncthreads();
  }

  const float sc = 1.0f / (float)N;
  for (int e = t; e < N; e += 512) {
    re[base + e] = sr[e] * sc;
    im[base + e] = si[e] * sc;
  }
}

// ---------------------------------------------------------------------------
// 32-point inverse FFT along c' (stride-N columns) + abs, fully in registers.
// Writes |ifft| into the `re` plane in place (per-thread read-before-write).
// Applies the 1/32 normalization (1/16384 already applied above).
// ---------------------------------------------------------------------------
__global__ void __launch_bounds__(256)
ifft32_cols_abs(float* re, float* im) {
  size_t idx = (size_t)blockIdx.x * 256 + threadIdx.x;  // B*HD*N threads
  int bh = (int)(idx >> 14);
  int n  = (int)(idx & (N - 1));
  size_t base = (size_t)bh * CPH * N + n;

  float xr[32], xi[32];
#pragma unroll
  for (int i = 0; i < 32; ++i) {
    int r = ((i & 1) << 4) | ((i & 2) << 2) | (i & 4) | ((i & 8) >> 2) |
            ((i & 16) >> 4);  // 5-bit reversal, folds at compile time
    xr[r] = re[base + (size_t)i * N];
    xi[r] = im[base + (size_t)i * N];
  }
#pragma unroll
  for (int s = 1; s <= 5; ++s) {
    int m = 1 << s, mh = m >> 1;
#pragma unroll
    for (int k = 0; k < 32; k += (1 << s)) {
#pragma unroll
      for (int j = 0; j < (1 << (s - 1)); ++j) {
        float ang = TWO_PI * (float)j / (float)m;
        float cw, sw;
        __sincosf(ang, &sw, &cw);
        int p = k + j;
        float vr = xr[p + mh], vi = xi[p + mh];
        float tr = vr * cw - vi * sw;
        float ti = vr * sw + vi * cw;
        float ur = xr[p], ui = xi[p];
        xr[p]      = ur + tr;  xi[p]      = ui + ti;
        xr[p + mh] = ur - tr;  xi[p + mh] = ui - ti;
      }
    }
  }
  const float sc = 1.0f / 32.0f;
#pragma unroll
  for (int i = 0; i < 32; ++i) {
    float a = xr[i] * sc, b = xi[i] * sc;
    re[base + (size_t)i * N] = sqrtf(a * a + b * b);
  }
}

// ---------------------------------------------------------------------------
// Fused gating branch: conv1x1(256->16) + BN(inference) + ReLU +
// conv1x1(16->256) + sigmoid on xf.real, then gate*(xf.re, xf.im).
// Weights staged in LDS; one thread per spatial position.
// ---------------------------------------------------------------------------
__global__ void __launch_bounds__(256)
gating_kernel(const float* xr, const float* xi, const float* w1,
              const float* b1, const float* bn_g, const float* bn_b,
              const float* bn_m, const float* bn_v, const float* w2,
              const float* b2, float* gr, float* gi) {
  __shared__ float w1s[CR * C];   // 16KB
  __shared__ float w2s[C * CR];   // 16KB
  __shared__ float bnA[CR], bnB[CR], b2s[C];
  int t = threadIdx.x;
  for (int i = t; i < CR * C; i += 256) {
    w1s[i] = w1[i];
    w2s[i] = w2[i];
  }
  if (t < CR) {
    float iv = rsqrtf(bn_v[t] + 1e-5f);
    float a = bn_g[t] * iv;
    bnA[t] = a;
    bnB[t] = bn_b[t] + a * (b1[t] - bn_m[t]);
  }
  b2s[t] = b2[t];
  __syncthreads();

  size_t pos = (size_t)blockIdx.x * 256 + t;  // 0..B*N-1
  int b = (int)(pos >> 14);
  int hw = (int)(pos & (N - 1));
  size_t xbase = (size_t)b * C * N + hw;

  float acc[CR];
#pragma unroll
  for (int j = 0; j < CR; ++j) acc[j] = 0.0f;
  for (int c = 0; c < C; ++c) {
    float v = xr[xbase + (size_t)c * N];
#pragma unroll
    for (int j = 0; j < CR; ++j) acc[j] = fmaf(w1s[j * C + c], v, acc[j]);
  }
  float y[CR];
#pragma unroll
  for (int j = 0; j < CR; ++j) {
    float u = bnA[j] * acc[j] + bnB[j];
    y[j] = u > 0.0f ? u : 0.0f;
  }
  for (int o = 0; o < C; ++o) {
    float s = b2s[o];
#pragma unroll
    for (int j = 0; j < CR; ++j) s = fmaf(w2s[o * CR + j], y[j], s);
    float gate = 1.0f / (1.0f + __expf(-s));
    size_t ix = xbase + (size_t)o * N;
    gr[ix] = gate * xr[ix];
    gi[ix] = gate * xi[ix];
  }
}

// ---------------------------------------------------------------------------
// Final projection: out[b,o,hw] = sum_{c<512} cat[b,c,hw] * proj_w[o,c]
// (cat = [out_f ; out_f_l]).  f32 WMMA, one 16x16 tile per wave.
// ---------------------------------------------------------------------------
__global__ void __launch_bounds__(128)
proj_kernel(const float* outF, const float* outFL, const float* pw,
            float* dst) {
  int t = threadIdx.x;
  int wid = t >> 5, lane = t & 31;
  int o0 = blockIdx.x * 16;
  int n0 = blockIdx.y * 64 + wid * 16;
  int b = blockIdx.z;
  int r = lane & 15, half = lane >> 4;

  const float* aRow = pw + (size_t)(o0 + r) * (2 * C) + 2 * half;
  size_t bbase = (size_t)b * C * N;

  v8f acc = {};
  for (int k = 0; k < 2 * C; k += 4) {
    if ((k & 63) == 0) __builtin_prefetch(aRow + k + 64, 0, 1);
    v2f a = {aRow[k], aRow[k + 1]};
    int kb = k + 2 * half;
    const float* p1 = (kb < C) ? outF + bbase + (size_t)kb * N
                               : outFL + bbase + (size_t)(kb - C) * N;
    int kb2 = kb + 1;
    const float* p2 = (kb2 < C) ? outF + bbase + (size_t)kb2 * N
                                : outFL + bbase + (size_t)(kb2 - C) * N;
    v2f bb = {p1[n0 + r], p2[n0 + r]};
    acc = wmma_f32(a, bb, acc);
  }
#pragma unroll
  for (int rr = 0; rr < 8; ++rr) {
    int row = o0 + (half ? rr + 8 : rr);
    dst[((size_t)b * C + row) * N + n0 + (lane & 15)] = acc[rr];
  }
}

// ---------------------------------------------------------------------------
// Host-side orchestration.  Workspace layout (floats):
//   [0,P)   xf_re      -> later reused for out_f_l
//   [P,2P)  xf_im
//   [2P,3P) out_re     -> becomes out_f in place
//   [3P,4P) out_im     -> reused as gated_re
//   [4P,5P) gated_im
//   [5P, +32K) attn_re, [+32K] attn_im, [+64K] invn (1024)
// Total ~320MB + 0.3MB.
// ---------------------------------------------------------------------------
extern "C" void kernel_launch(void* const* d_in, const int* in_sizes, int n_in,
                              void* d_out, int out_size, void* d_ws,
                              size_t ws_size, hipStream_t stream) {
  const float* x           = (const float*)d_in[0];
  const float* temperature = (const float*)d_in[1];
  const float* w1          = (const float*)d_in[2];
  const float* b1          = (const float*)d_in[3];
  const float* bn_gamma    = (const float*)d_in[4];
  const float* bn_beta     = (const float*)d_in[5];
  const float* bn_mean     = (const float*)d_in[6];
  const float* bn_var      = (const float*)d_in[7];
  const float* w2          = (const float*)d_in[8];
  const float* b2          = (const float*)d_in[9];
  const float* proj_w      = (const float*)d_in[10];

  float* ws = (float*)d_ws;
  const size_t P = PLANE;
  float* xr = ws;
  float* xi = ws + P;
  float* oR = ws + 2 * P;
  float* oI = ws + 3 * P;     // later: gated_re
  float* g2 = ws + 4 * P;     // gated_im
  float* attnR = ws + 5 * P;
  float* attnI = attnR + (size_t)B * HD * 1024;
  float* invn  = attnI + (size_t)B * HD * 1024;

  const int lines = B * C * H;  // 131072

  // 1) forward fft2(x): rows then cols
  fft128_rows_r2c<<<lines, 64, 0, stream>>>(x, xr, xi);
  fft128_c2c<<<lines, 64, 0, stream>>>(xr, xi, 128, 1, 128, -1.0f, 1.0f,
                                       nullptr);
  // 2) per-channel inverse norms
  norm_kernel<<<B * C, 256, 0, stream>>>(xr, xi, invn);
  // 3) Gram + complex softmax -> attn (TDM double-buffered)
  gram_attn_kernel<<<B * HD, 128, 0, stream>>>(xr, xi, invn, temperature,
                                               attnR, attnI);
  // 4) out = attn @ qkv (TDM-staged B tiles)
  apply_attn_kernel<<<dim3(B * HD, N / 32), 128, 0, stream>>>(xr, xi, attnR,
                                                              attnI, oR, oI);
  // 5) ifft2 over (c'=32, n=16384), then abs -> out_f (in place in oR)
  ifft16384_rows<<<B * C, 512, 2 * N * sizeof(float), stream>>>(oR, oI);
  ifft32_cols_abs<<<(B * HD * N) / 256, 256, 0, stream>>>(oR, oI);
  // 6) gating branch -> gated planes (oI, g2)
  gating_kernel<<<(B * N) / 256, 256, 0, stream>>>(
      xr, xi, w1, b1, bn_gamma, bn_beta, bn_mean, bn_var, w2, b2, oI, g2);
  // 7) standard ifft2 over (h,w): cols then rows (+abs) -> out_f_l in xr
  fft128_c2c<<<lines, 64, 0, stream>>>(oI, g2, 128, 1, 128, 1.0f,
                                       1.0f / 128.0f, nullptr);
  fft128_c2c<<<lines, 64, 0, stream>>>(oI, g2, 128, 128, 1, 1.0f,
                                       1.0f / 128.0f, xr);
  // 8) projection of concat -> d_out
  proj_kernel<<<dim3(C / 16, N / 64, B), 128, 0, stream>>>(oR, xr, proj_w,
                                                           (float*)d_out);
}